// MultiBoxLoss_69853348102904
// MI455X (gfx1250) — compile-verified
//
#include <hip/hip_runtime.h>
#include <math.h>

#define NB 128
#define NP 8732
#define NC 21
#define NM 16
#define SORT_N 16384   // next pow2 >= NP

typedef __attribute__((ext_vector_type(2))) float v2f;
typedef __attribute__((ext_vector_type(8))) float v8f;
typedef __attribute__((ext_vector_type(4))) int   v4i;

// gfx1250 async global->LDS copy (ASYNCcnt-tracked), guarded so absence of the
// builtin falls back to plain loads. Prototype (from clang diagnostic):
//   void (v4i addrspace(1)*, v4i addrspace(3)*, imm int offset, imm int cpol)
#if defined(__has_builtin)
#if __has_builtin(__builtin_amdgcn_global_load_async_to_lds_b128)
#define HAVE_ASYNC_LDS 1
#endif
#endif
#ifndef HAVE_ASYNC_LDS
#define HAVE_ASYNC_LDS 0
#endif

typedef __attribute__((address_space(1))) v4i g_v4i;   // global b128 chunk
typedef __attribute__((address_space(3))) v4i l_v4i;   // LDS b128 chunk

// ---------------------------------------------------------------------------
// Wave32 sum via the matrix pipe: one v_wmma_f32_16x16x4_f32 with all-ones B
// reduces 2 values per lane (64 values) to row sums; since B is all ones every
// D column is identical, so summing the 8 D regs per lane gives the half-wave
// total, and one xor-16 shuffle completes the full wave sum in every lane.
// Requires EXEC all ones -> only call from uniform control flow, full blocks.
// ---------------------------------------------------------------------------
__device__ __forceinline__ float wave_sum32(float x) {
  v2f a;  a[0] = x;    a[1] = 0.0f;
  v2f bo; bo[0] = 1.0f; bo[1] = 1.0f;
  v8f c = {};
  v8f d = __builtin_amdgcn_wmma_f32_16x16x4_f32(false, a, false, bo,
                                                (short)0, c, false, false);
  float s = d[0] + d[1] + d[2] + d[3] + d[4] + d[5] + d[6] + d[7];
  s += __shfl_xor(s, 16, 32);
  return s;
}

__device__ __forceinline__ float block_sum(float x, float* red) {
  const int lane = threadIdx.x & 31;
  const int wave = threadIdx.x >> 5;
  const int nw   = blockDim.x >> 5;
  float ws = wave_sum32(x);
  if (lane == 0) red[wave] = ws;
  __syncthreads();
  if (threadIdx.x == 0) {
    float t = 0.0f;
    for (int w = 0; w < nw; ++w) t += red[w];
    red[0] = t;
  }
  __syncthreads();
  float tot = red[0];
  __syncthreads();
  return tot;
}

// ---------------------------------------------------------------------------
// Kernel 1: per-image matching + encoded loc targets + smooth-L1 loc loss.
// One block per image, 256 threads.
// ---------------------------------------------------------------------------
__global__ __launch_bounds__(256)
void match_kernel(const float* __restrict__ loc_data,
                  const float* __restrict__ priors,
                  const float* __restrict__ targets,
                  int*   __restrict__ conf_t,
                  float* __restrict__ bto,   // [B,P] best-truth-overlap (scratch)
                  int*   __restrict__ bti,   // [B,P] best-truth-idx    (scratch)
                  float* __restrict__ loss_l_img,
                  int*   __restrict__ num_pos) {
  __shared__ float tx0[NM], ty0[NM], tx1[NM], ty1[NM], tlab[NM], tarea[NM];
  __shared__ unsigned long long bestPack[NM];   // per-truth argmax over priors
  __shared__ float red[32];

  const int b   = blockIdx.x;
  const int tid = threadIdx.x;

  if (tid < NM) {
    const float* t = targets + (size_t)(b * NM + tid) * 5;
    tx0[tid] = t[0]; ty0[tid] = t[1]; tx1[tid] = t[2]; ty1[tid] = t[3];
    tlab[tid] = t[4];
    tarea[tid] = (t[2] - t[0]) * (t[3] - t[1]);
    bestPack[tid] = 0ull;
  }
  __syncthreads();

  float    tmax[NM];
  unsigned tpr[NM];
#pragma unroll
  for (int j = 0; j < NM; ++j) { tmax[j] = -1.0f; tpr[j] = 0u; }

  const float4* pr4  = (const float4*)priors;
  float*        btoB = bto + (size_t)b * NP;
  int*          btiB = bti + (size_t)b * NP;

  for (int p = tid; p < NP; p += 256) {
    float4 pc = pr4[p];
    float px0 = pc.x - pc.z * 0.5f, py0 = pc.y - pc.w * 0.5f;
    float px1 = pc.x + pc.z * 0.5f, py1 = pc.y + pc.w * 0.5f;
    float areab = pc.z * pc.w;
    float bov = -1.0f; int bj = 0;
#pragma unroll
    for (int j = 0; j < NM; ++j) {
      float lx = fmaxf(tx0[j], px0), ly = fmaxf(ty0[j], py0);
      float rx = fminf(tx1[j], px1), ry = fminf(ty1[j], py1);
      float iw = fmaxf(rx - lx, 0.0f), ih = fmaxf(ry - ly, 0.0f);
      float inter = iw * ih;
      float iou = inter / (tarea[j] + areab - inter);
      if (iou > bov)     { bov = iou; bj = j; }           // first-idx tie-break
      if (iou > tmax[j]) { tmax[j] = iou; tpr[j] = (unsigned)p; }
    }
    btoB[p] = bov;
    btiB[p] = bj;
  }

  // per-truth argmax over priors: packed (overlap_bits << 32) | ~prior_idx
  // so ties resolve to the smallest prior index (jnp.argmax semantics).
#pragma unroll
  for (int j = 0; j < NM; ++j) {
    unsigned long long pk =
        ((unsigned long long)__float_as_uint(fmaxf(tmax[j], 0.0f)) << 32) |
        (unsigned long long)(0xFFFFFFFFu - tpr[j]);
    atomicMax(&bestPack[j], pk);   // ds_max_rtn_u64 on LDS
  }
  __syncthreads();

  if (tid == 0) {   // forced matches, sequential last-wins like the reference
    for (int j = 0; j < NM; ++j) {
      unsigned p = 0xFFFFFFFFu - (unsigned)(bestPack[j] & 0xFFFFFFFFull);
      btoB[p] = 2.0f;
      btiB[p] = j;
    }
  }
  __syncthreads();

  float acc = 0.0f;
  int   npos = 0;
  const float4* loc4 = (const float4*)loc_data + (size_t)b * NP;
  for (int p = tid; p < NP; p += 256) {
    float ov = btoB[p];
    int   j  = btiB[p];
    int   cf = (ov < 0.5f) ? 0 : ((int)tlab[j] + 1);
    conf_t[(size_t)b * NP + p] = cf;
    if (cf > 0) {
      float4 pc = pr4[p];
      float mx0 = tx0[j], my0 = ty0[j], mx1 = tx1[j], my1 = ty1[j];
      float gx = ((mx0 + mx1) * 0.5f - pc.x) / (0.1f * pc.z);
      float gy = ((my0 + my1) * 0.5f - pc.y) / (0.1f * pc.w);
      float gw = logf((mx1 - mx0) / pc.z) / 0.2f;
      float gh = logf((my1 - my0) / pc.w) / 0.2f;
      float4 ld = loc4[p];
      float d0 = fabsf(ld.x - gx), d1 = fabsf(ld.y - gy);
      float d2 = fabsf(ld.z - gw), d3 = fabsf(ld.w - gh);
      acc += (d0 < 1.f ? 0.5f * d0 * d0 : d0 - 0.5f)
           + (d1 < 1.f ? 0.5f * d1 * d1 : d1 - 0.5f)
           + (d2 < 1.f ? 0.5f * d2 * d2 : d2 - 0.5f)
           + (d3 < 1.f ? 0.5f * d3 * d3 : d3 - 0.5f);
      npos++;
    }
  }
  float lsum = block_sum(acc, red);
  float psum = block_sum((float)npos, red);
  if (tid == 0) { loss_l_img[b] = lsum; num_pos[b] = (int)(psum + 0.5f); }
}

// ---------------------------------------------------------------------------
// Kernel 2: per-prior CE = logsumexp - gathered logit; mining = CE for negs.
// 4 rows per thread: 4 rows * 21 floats = 336 B = 21 x float4, so the dominant
// 94 MB conf_data stream moves with global_load_b128 instead of 21 dword loads
// per row. Gather is done with unrolled compares (v_cndmask), never a dynamic
// register index (which would spill to scratch).
// ---------------------------------------------------------------------------
#define NROWS ((size_t)NB * NP)   // divisible by 4 (NP = 4*2183)

__global__ __launch_bounds__(256)
void ce_kernel(const float* __restrict__ conf_data,
               const int*   __restrict__ conf_t,
               float* __restrict__ ce,
               float* __restrict__ mining) {
  size_t q = (size_t)blockIdx.x * blockDim.x + threadIdx.x;  // 4-row group id
  if (q * 4 >= NROWS) return;
  const float4* g = (const float4*)conf_data + q * NC;       // 21 float4 = 4 rows
  __builtin_prefetch((const float*)(g + 256 * NC), 0, 1);    // global_prefetch_b8

  float v[4 * NC];
#pragma unroll
  for (int k = 0; k < NC; ++k) {
    float4 t = g[k];
    v[4 * k + 0] = t.x; v[4 * k + 1] = t.y;
    v[4 * k + 2] = t.z; v[4 * k + 3] = t.w;
  }

#pragma unroll
  for (int r = 0; r < 4; ++r) {
    size_t i = q * 4 + r;
    int ct = conf_t[i];
    float m = v[NC * r];
#pragma unroll
    for (int c = 1; c < NC; ++c) m = fmaxf(m, v[NC * r + c]);
    float se = 0.0f;
    float gathered = 0.0f;
#pragma unroll
    for (int c = 0; c < NC; ++c) {
      float x = v[NC * r + c];
      se += expf(x - m);
      gathered = (c == ct) ? x : gathered;
    }
    float cev = m + logf(se) - gathered;
    ce[i] = cev;
    mining[i] = (ct > 0) ? 0.0f : cev;
  }
}

// ---------------------------------------------------------------------------
// Kernel 3: per-image hard-negative mining. The 34 KB mining slice is pulled
// into LDS with GLOBAL_LOAD_ASYNC_TO_LDS_B128 (ASYNCcnt), overlapping with the
// positive-CE global-read pass; then bitonic sort (descending) + top-k sum.
// ---------------------------------------------------------------------------
__global__ __launch_bounds__(1024)
void mine_kernel(const float* __restrict__ ce,
                 const float* __restrict__ mining,
                 const int*   __restrict__ conf_t,
                 const int*   __restrict__ num_pos,
                 float* __restrict__ loss_c_img) {
  __shared__ __align__(16) float s[SORT_N];   // 64 KB of the WGP's 320 KB LDS
  __shared__ float red[32];

  const int b   = blockIdx.x;
  const int tid = threadIdx.x;
  const size_t base = (size_t)b * NP;   // 16B-aligned: b*8732*4 = b*16*2183

#if HAVE_ASYNC_LDS
  {
    g_v4i* gsrc = (g_v4i*)(mining + base);   // addrspace(1) b128 chunks
    l_v4i* ldst = (l_v4i*)s;                 // addrspace(3) b128 chunks
    for (int i4 = tid; i4 < NP / 4; i4 += 1024)   // NP/4 = 2183 exactly
      __builtin_amdgcn_global_load_async_to_lds_b128(gsrc + i4, ldst + i4, 0, 0);
  }
#else
  for (int i = tid; i < NP; i += 1024) s[i] = mining[base + i];
#endif
  for (int i = NP + tid; i < SORT_N; i += 1024) s[i] = -1.0f;  // CE>=0; pad sinks

  // overlap: positive-CE sum reads globals while async DMA fills LDS
  float pacc = 0.0f;
  for (int i = tid; i < NP; i += 1024)
    if (conf_t[base + i] > 0) pacc += ce[base + i];

#if HAVE_ASYNC_LDS
  asm volatile("s_wait_asynccnt 0x0" ::: "memory");
#endif
  __syncthreads();

  float posce = block_sum(pacc, red);
  __syncthreads();

  // bitonic sort, descending overall
  for (unsigned k = 2; k <= SORT_N; k <<= 1) {
    for (unsigned j = k >> 1; j > 0; j >>= 1) {
      for (unsigned i = tid; i < SORT_N; i += 1024) {
        unsigned ixj = i ^ j;
        if (ixj > i) {
          float a = s[i], c2 = s[ixj];
          bool desc = ((i & k) == 0);
          if (desc ? (a < c2) : (a > c2)) { s[i] = c2; s[ixj] = a; }
        }
      }
      __syncthreads();
    }
  }

  int nn = num_pos[b] * 3;           // NEGPOS * num_pos, clipped to P-1
  if (nn > NP - 1) nn = NP - 1;
  float tacc = 0.0f;
  for (int i = tid; i < nn; i += 1024) tacc += s[i];
  float topk = block_sum(tacc, red);
  if (tid == 0) loss_c_img[b] = posce + topk;
}

// ---------------------------------------------------------------------------
// Kernel 4: final cross-image reduction, out = (loss_l/N, loss_c/N).
// ---------------------------------------------------------------------------
__global__ __launch_bounds__(128)
void final_kernel(const float* __restrict__ loss_l_img,
                  const float* __restrict__ loss_c_img,
                  const int*   __restrict__ num_pos,
                  float* __restrict__ out) {
  __shared__ float red[32];
  const int tid = threadIdx.x;   // 128 threads == NB, all active
  float L  = block_sum(loss_l_img[tid], red);
  float Cc = block_sum(loss_c_img[tid], red);
  float N  = block_sum((float)num_pos[tid], red);
  if (tid == 0) { out[0] = L / N; out[1] = Cc / N; }
}

// ---------------------------------------------------------------------------
extern "C" void kernel_launch(void* const* d_in, const int* in_sizes, int n_in,
                              void* d_out, int out_size, void* d_ws, size_t ws_size,
                              hipStream_t stream) {
  const float* loc_data  = (const float*)d_in[0];  // [B,P,4]
  const float* conf_data = (const float*)d_in[1];  // [B,P,C]
  const float* priors    = (const float*)d_in[2];  // [P,4]
  const float* targets   = (const float*)d_in[3];  // [B*M,5]
  // d_in[4] targets_idx: fixed contiguous layout, unused

  const size_t np = (size_t)NB * NP;
  // workspace layout (regions reused across phases):
  int*   conf_t = (int*)d_ws;                    // [B,P] int
  float* r1     = (float*)(conf_t + np);         // bto (K1) -> ce (K2/K3)
  int*   r2     = (int*)(r1 + np);               // bti (K1) -> mining (K2/K3)
  float* small_ = (float*)(r2 + np);
  float* loss_l_img = small_;                    // [B]
  float* loss_c_img = small_ + NB;               // [B]
  int*   num_pos    = (int*)(small_ + 2 * NB);   // [B]
  float* ce     = r1;
  float* mining = (float*)r2;

  match_kernel<<<NB, 256, 0, stream>>>(loc_data, priors, targets,
                                       conf_t, r1, r2, loss_l_img, num_pos);
  const int ngrp = (int)((np / 4 + 255) / 256);
  ce_kernel<<<ngrp, 256, 0, stream>>>(conf_data, conf_t, ce, mining);
  mine_kernel<<<NB, 1024, 0, stream>>>(ce, mining, conf_t, num_pos, loss_c_img);
  final_kernel<<<1, 128, 0, stream>>>(loss_l_img, loss_c_img, num_pos,
                                      (float*)d_out);
}